// Quantizer_33139967656119
// MI455X (gfx1250) — compile-verified
//
#include <hip/hip_runtime.h>

typedef __attribute__((ext_vector_type(2))) float v2f;
typedef __attribute__((ext_vector_type(8))) float v8f;

#define EMBED_DIM 256
#define KSTEPS 64   // 256 / 4 (K-slab per fp32 WMMA)

// ---------------------------------------------------------------------------
// Kernel E: enorm[c] = ||embed[c]||^2  (1 MB read, trivial)
// ---------------------------------------------------------------------------
__global__ __launch_bounds__(256) void vq_enorm_kernel(
    const float* __restrict__ embed, float* __restrict__ enorm, int K)
{
    int c = blockIdx.x * blockDim.x + threadIdx.x;
    if (c >= K) return;
    const float4* row = (const float4*)(embed + (size_t)c * EMBED_DIM);
    float s = 0.f;
#pragma unroll 8
    for (int i = 0; i < EMBED_DIM / 4; ++i) {
        float4 v = row[i];
        s += v.x * v.x + v.y * v.y + v.z * v.z + v.w * v.w;
    }
    enorm[c] = s;
}

// ---------------------------------------------------------------------------
// Kernel A: per-token argmin_k (||e_k||^2 - 2 x.e_k) using fp32 WMMA 16x16x4.
// One wave owns 16 tokens. A panel (16 tok x 256 dim) lives in 128 VGPRs for
// the whole kernel; B fragments stream from the L2-resident codebook.
//
// f32 fragment layout (ISA 7.12.2): value(outer = lane&15,
//                                         k = 4*s + 2*(lane>>4) + {v0,v1})
// -> both A and B fragments are contiguous float2 loads from row-major data.
// C layout: vgpr r, lane l: token M = r + 8*(l>>4), code N = l&15.
// ---------------------------------------------------------------------------
__global__ __launch_bounds__(256) void vq_argmin_kernel(
    const float* __restrict__ x, const float* __restrict__ embed,
    const float* __restrict__ enorm, int* __restrict__ idxOut, int numCodeTiles)
{
    const int lane = threadIdx.x & 31;
    const int wave = threadIdx.x >> 5;
    const int lm   = lane & 15;
    const int hh   = lane >> 4;   // which 16-lane half
    const int tokBase = (blockIdx.x * (blockDim.x >> 5) + wave) << 4;

    // Load this wave's full A panel into registers (one-time, 64 MB total).
    const v2f* arow = (const v2f*)(x + (size_t)(tokBase + lm) * EMBED_DIM);
    v2f a[KSTEPS];
#pragma unroll
    for (int s = 0; s < KSTEPS; ++s) a[s] = arow[2 * s + hh];

    // Running per-lane best (this lane's code column), per token register r.
    float bestS[8];
    int   bestI[8];
#pragma unroll
    for (int r = 0; r < 8; ++r) { bestS[r] = __builtin_inff(); bestI[r] = 0x7fffffff; }

#pragma unroll 1
    for (int j = 0; j < numCodeTiles; ++j) {
        const int code = (j << 4) + lm;
        const v2f* brow = (const v2f*)(embed + (size_t)code * EMBED_DIM);

        v8f acc = {};
#pragma unroll
        for (int s = 0; s < KSTEPS; ++s) {
            v2f b = brow[2 * s + hh];
            // D = A x B + C, full fp32 (V_WMMA_F32_16X16X4_F32)
            acc = __builtin_amdgcn_wmma_f32_16x16x4_f32(
                false, a[s], false, b, (short)0, acc, false, false);
        }

        const float en = enorm[code];
#pragma unroll
        for (int r = 0; r < 8; ++r) {
            float score = __builtin_fmaf(-2.0f, acc[r], en);
            // strict < keeps the earliest (lowest) code index within this lane
            if (score < bestS[r]) { bestS[r] = score; bestI[r] = code; }
        }
    }

    // Final cross-lane argmin (within each 16-lane half), tie-break lowest idx
    // to match jnp.argmin's first-minimum semantics.
#pragma unroll
    for (int r = 0; r < 8; ++r) {
        float bs = bestS[r];
        int   bi = bestI[r];
#pragma unroll
        for (int m = 8; m >= 1; m >>= 1) {
            float os = __shfl_xor(bs, m, 32);
            int   oi = __shfl_xor(bi, m, 32);
            if (os < bs || (os == bs && oi < bi)) { bs = os; bi = oi; }
        }
        if (lm == r) idxOut[tokBase + (hh << 3) + r] = bi;
    }
}

// ---------------------------------------------------------------------------
// Kernel B: quantized gather + straight-through output + per-block SSE.
// Block = 256 threads = 4 tokens x 64 float4. Deterministic LDS tree reduce.
// ---------------------------------------------------------------------------
__global__ __launch_bounds__(256) void vq_gather_loss_kernel(
    const float* __restrict__ x, const float* __restrict__ embed,
    const int* __restrict__ idx, float* __restrict__ out,
    float* __restrict__ partial)
{
    const int tid = threadIdx.x;
    const int tok = blockIdx.x * 4 + (tid >> 6);
    const int d4  = tid & 63;
    const int code = idx[tok];

    float4 q  = ((const float4*)(embed + (size_t)code * EMBED_DIM))[d4];
    float4 xi = ((const float4*)(x + (size_t)tok * EMBED_DIM))[d4];

    float4 d;
    d.x = q.x - xi.x; d.y = q.y - xi.y; d.z = q.z - xi.z; d.w = q.w - xi.w;
    // straight-through: out = x + (q - x), reproducing the reference's fp ops
    float4 o;
    o.x = xi.x + d.x; o.y = xi.y + d.y; o.z = xi.z + d.z; o.w = xi.w + d.w;
    ((float4*)(out + (size_t)tok * EMBED_DIM))[d4] = o;

    float s = d.x * d.x + d.y * d.y + d.z * d.z + d.w * d.w;

    __shared__ float red[256];
    red[tid] = s;
    __syncthreads();
#pragma unroll
    for (int off = 128; off >= 1; off >>= 1) {
        if (tid < off) red[tid] += red[tid + off];
        __syncthreads();
    }
    if (tid == 0) partial[blockIdx.x] = red[0];
}

// ---------------------------------------------------------------------------
// Kernel C: deterministic final reduction -> latent_loss at d_out[N*D].
// latent_loss = (1 + 0.25) * mean((q - x)^2)
// ---------------------------------------------------------------------------
__global__ __launch_bounds__(256) void vq_loss_finalize_kernel(
    const float* __restrict__ partial, int numPartials,
    float* __restrict__ lossOut, float invCount)
{
    const int tid = threadIdx.x;
    float s = 0.f;
    for (int i = tid; i < numPartials; i += 256) s += partial[i];
    __shared__ float red[256];
    red[tid] = s;
    __syncthreads();
#pragma unroll
    for (int off = 128; off >= 1; off >>= 1) {
        if (tid < off) red[tid] += red[tid + off];
        __syncthreads();
    }
    if (tid == 0) *lossOut = 1.25f * red[0] * invCount;
}

// ---------------------------------------------------------------------------
extern "C" void kernel_launch(void* const* d_in, const int* in_sizes, int n_in,
                              void* d_out, int out_size, void* d_ws, size_t ws_size,
                              hipStream_t stream)
{
    const float* x     = (const float*)d_in[0];   // [32,2048,256] fp32
    const float* embed = (const float*)d_in[1];   // [1024,256]    fp32
    float* out = (float*)d_out;                   // [N*D quantized] + [1 loss]

    const int ND = in_sizes[0];            // 16,777,216
    const int N  = ND / EMBED_DIM;         // 65,536 tokens
    const int K  = in_sizes[1] / EMBED_DIM; // 1,024 codes

    // workspace layout (all fully overwritten every call -> no init needed)
    char* ws = (char*)d_ws;
    float* enorm   = (float*)ws;                                   // K floats
    int*   idx     = (int*)(ws + (((size_t)K * 4 + 255) & ~255u)); // N ints
    float* partial = (float*)((char*)idx + (((size_t)N * 4 + 255) & ~255u)); // N/4 floats

    // E: code norms
    vq_enorm_kernel<<<(K + 255) / 256, 256, 0, stream>>>(embed, enorm, K);

    // A: fp32-WMMA argmin. 256 thr = 8 waves = 128 tokens per block.
    vq_argmin_kernel<<<N / 128, 256, 0, stream>>>(x, embed, enorm, idx, K / 16);

    // B: gather + STE + partial SSE (4 tokens per block)
    const int nBlocksB = N / 4;
    vq_gather_loss_kernel<<<nBlocksB, 256, 0, stream>>>(x, embed, idx, out, partial);

    // C: finalize scalar loss into the last output element
    vq_loss_finalize_kernel<<<1, 256, 0, stream>>>(
        partial, nBlocksB, out + (size_t)ND, 1.0f / (float)ND);
}